// FlowmatchingActionHead_2723009266342
// MI455X (gfx1250) — compile-verified
//
#include <hip/hip_runtime.h>
#include <hip/hip_bf16.h>

typedef __attribute__((ext_vector_type(2))) float v2f;
typedef __attribute__((ext_vector_type(8))) float v8f;

#define HID   1536
#define NB    32      // batch
#define NT    32      // rows M per sample
#define AD    32      // action dim
#define KC    32      // K-chunk staged in LDS
#define TN    128     // N tile per workgroup
#define ASTR  36      // padded LDS row stride for A (16B aligned, conflict-free)
#define LOG1E4 9.210340372f

// LDS byte offset = low 32 bits of the generic address (LDS aperture rule).
__device__ __forceinline__ uint32_t lds_off(const void* p) {
    return (uint32_t)(uintptr_t)p;
}

// gfx1250 async bulk copy global -> LDS (16B per lane), tracked on ASYNCcnt.
__device__ __forceinline__ void async_copy_b128(uint32_t lds_byte, const float* gaddr) {
    asm volatile("global_load_async_to_lds_b128 %0, %1, off"
                 :: "v"(lds_byte), "v"((uint64_t)(uintptr_t)gaddr)
                 : "memory");
}

__device__ __forceinline__ void wait_async_5() {
    asm volatile("s_wait_asynccnt 0x5" ::: "memory");
}
__device__ __forceinline__ void wait_async_0() {
    asm volatile("s_wait_asynccnt 0x0" ::: "memory");
}

// ---------------------------------------------------------------------------
// Per-sample (categorical) GEMM:
//   out[b, m, n] = act( sum_k X[b,m,k] * W[cat[b]][k,n] + bias[cat[b]][n] )
// M = 32 rows. WG = 256 threads (8 wave32), tile 32(M) x 128(N), K-chunked by
// 32 with double-buffered async global->LDS staging. Each wave owns a 16x32
// slab = two 16x16 V_WMMA_F32_16X16X4_F32 accumulators.
// ---------------------------------------------------------------------------
template <bool SWISH>
__global__ __launch_bounds__(256)
void gemm_cat_kernel(const float* __restrict__ X, int ldx, int K,
                     const float* __restrict__ W,     // [C, K, HID]
                     const float* __restrict__ bias,  // [C, HID]
                     int strideWc, int strideBc,
                     const int* __restrict__ cat_ids,
                     float* __restrict__ out, int ldo, int out_col0,
                     int sampleStrideX, int sampleStrideOut)
{
    __shared__ float shW[2][KC][TN];    // [buf][k][n]  32 KB
    __shared__ float shA[2][NT][ASTR];  // [buf][m][k]   9 KB

    const int b = blockIdx.y;
    const int c = cat_ids[b];
    const float* __restrict__ Xb = X + (size_t)b * sampleStrideX;
    const float* __restrict__ Wc = W + (size_t)c * strideWc;
    const float* __restrict__ bc = bias + (size_t)c * strideBc;
    float* __restrict__ outb = out + (size_t)b * sampleStrideOut + out_col0;

    const int tid    = threadIdx.x;
    const int lane   = tid & 31;
    const int wave   = tid >> 5;
    const int g      = lane >> 4;      // lane group
    const int l      = lane & 15;
    const int mbase  = (wave & 1) * 16;
    const int nq     = (wave >> 1) * 32;       // wave's col base within tile
    const int nblock = blockIdx.x * TN;        // tile's col base in [0,HID)

    // Stage one K-chunk (32 x 128 of W, 32 x 32 of X) into LDS buffer `buf`.
    // Every wave issues exactly 5 async-load instructions per call.
    auto stage = [&](int buf, int kbase) {
        #pragma unroll
        for (int i = 0; i < 4; ++i) {
            const int idx = tid + i * 256;          // 0..1023
            const int row = idx >> 5;               // 0..31 (k row)
            const int c4  = (idx & 31) << 2;        // 0..124 (n col, x4 floats)
            async_copy_b128(lds_off(&shW[buf][row][c4]),
                            Wc + (size_t)(kbase + row) * HID + nblock + c4);
        }
        {
            const int row = tid >> 3;               // 0..31 (m row)
            const int c4  = (tid & 7) << 2;         // 0..28 (k col, x4 floats)
            async_copy_b128(lds_off(&shA[buf][row][c4]),
                            Xb + (size_t)row * ldx + kbase + c4);
        }
    };

    v8f acc0 = {};   // cols [nblock+nq,    +16)
    v8f acc1 = {};   // cols [nblock+nq+16, +32)

    const int nchunks = K / KC;
    stage(0, 0);
    int buf = 0;

    for (int ch = 0; ch < nchunks; ++ch) {          // uniform control flow
        if (ch + 1 < nchunks) {
            stage(buf ^ 1, (ch + 1) * KC);
            wait_async_5();                          // chunk `ch` complete
        } else {
            wait_async_0();
        }
        __syncthreads();

        #pragma unroll
        for (int kk = 0; kk < KC; kk += 4) {
            // A 16x4: lanes 0-15 K=kk,kk+1 ; lanes 16-31 K=kk+2,kk+3
            v2f a = *(const v2f*)&shA[buf][mbase + l][kk + 2 * g];
            // B 4x16: VGPR0 row K=kk(+2g), VGPR1 next K row
            const float* w0 = &shW[buf][kk + 2 * g][0];
            const float* w1 = w0 + TN;
            v2f bm0, bm1;
            bm0.x = w0[nq + l];      bm0.y = w1[nq + l];
            bm1.x = w0[nq + 16 + l]; bm1.y = w1[nq + 16 + l];
            acc0 = __builtin_amdgcn_wmma_f32_16x16x4_f32(
                       false, a, false, bm0, (short)0, acc0, false, false);
            acc1 = __builtin_amdgcn_wmma_f32_16x16x4_f32(
                       false, a, false, bm1, (short)0, acc1, false, false);
        }
        __syncthreads();
        buf ^= 1;
    }

    // Epilogue. C/D 16x16 f32: VGPR r -> row mbase + r + 8*g, col = base + l.
    const int col0 = nblock + nq + l;
    const int col1 = col0 + 16;
    const float bias0 = bc[col0];
    const float bias1 = bc[col1];
    #pragma unroll
    for (int r = 0; r < 8; ++r) {
        const int row = mbase + r + 8 * g;
        float v0 = acc0[r] + bias0;
        float v1 = acc1[r] + bias1;
        if (SWISH) {
            v0 = v0 / (1.0f + __expf(-v0));
            v1 = v1 / (1.0f + __expf(-v1));
        }
        outb[(size_t)row * ldo + col0] = v0;
        outb[(size_t)row * ldo + col1] = v1;
    }
}

// ---------------------------------------------------------------------------
// Sinusoidal timestep embedding into concat buffer cols [HID, 2*HID).
// ---------------------------------------------------------------------------
__global__ __launch_bounds__(256)
void tau_kernel(const int* __restrict__ timesteps, float* __restrict__ x)
{
    const int b = blockIdx.x;
    const float ts = (float)timesteps[b];
    float* __restrict__ xb = x + (size_t)b * NT * (2 * HID) + HID;
    for (int j = threadIdx.x; j < HID; j += blockDim.x) {
        const int i = j >> 1;
        const float div = __expf(-LOG1E4 * (float)i * (1.0f / (float)(HID / 2)));
        const float arg = ts * div;
        const float v = (j & 1) ? __cosf(arg) : __sinf(arg);
        #pragma unroll 4
        for (int t = 0; t < NT; ++t)
            xb[(size_t)t * (2 * HID) + j] = v;
    }
}

// ---------------------------------------------------------------------------
extern "C" void kernel_launch(void* const* d_in, const int* in_sizes, int n_in,
                              void* d_out, int out_size, void* d_ws, size_t ws_size,
                              hipStream_t stream)
{
    const float* actions   = (const float*)d_in[0];   // (B,T,32)
    const int*   timesteps = (const int*)  d_in[1];   // (B,)
    const int*   cat_ids   = (const int*)  d_in[2];   // (B,)
    const float* W1        = (const float*)d_in[3];   // (32,32,1536)
    const float* b1        = (const float*)d_in[4];   // (32,1536)
    const float* W2        = (const float*)d_in[5];   // (32,3072,1536)
    const float* b2        = (const float*)d_in[6];   // (32,1536)
    const float* W3        = (const float*)d_in[7];   // (32,1536,1536)
    const float* b3        = (const float*)d_in[8];   // (32,1536)
    float* out = (float*)d_out;                       // (B,T,1536)

    float* x = (float*)d_ws;                          // (B,32,3072)
    float* h = x + (size_t)NB * NT * (2 * HID);       // (B,32,1536)

    dim3 blk(256);
    dim3 grid(HID / TN, NB);

    // 1) a_emb -> x[:, :, 0:HID]   (K = 32)
    hipLaunchKernelGGL((gemm_cat_kernel<false>), grid, blk, 0, stream,
        actions, AD, AD, W1, b1, AD * HID, HID, cat_ids,
        x, 2 * HID, 0, NT * AD, NT * 2 * HID);

    // 2) tau -> x[:, :, HID:2*HID]
    hipLaunchKernelGGL(tau_kernel, dim3(NB), blk, 0, stream, timesteps, x);

    // 3) h = swish(x @ W2 + b2)    (K = 3072)
    hipLaunchKernelGGL((gemm_cat_kernel<true>), grid, blk, 0, stream,
        x, 2 * HID, 2 * HID, W2, b2, 2 * HID * HID, HID, cat_ids,
        h, HID, 0, NT * 2 * HID, NT * HID);

    // 4) out = h @ W3 + b3         (K = 1536)
    hipLaunchKernelGGL((gemm_cat_kernel<false>), grid, blk, 0, stream,
        h, HID, HID, W3, b3, HID * HID, HID, cat_ids,
        out, HID, 0, NT * HID, NT * HID);
}